// MultiresolutionHashEncoding_44143673868849
// MI455X (gfx1250) — compile-verified
//
#include <hip/hip_runtime.h>
#include <math.h>

// Multiresolution hash encoding (Instant-NGP style), MI455X / gfx1250.
// L=16 levels, T=2^19 entries/level, D=3, F=2 features, N=500000 points.
// One thread per (point, level): 8 in-flight float2 gathers (L2-resident
// tables), trilinear interpolation, one coalesced non-temporal float2 store.

#define L_LV   16
#define T_SZ   524288u
#define T_MASK (T_SZ - 1u)
#define PRIME1 2654435761u
#define PRIME2 805459861u

typedef __attribute__((ext_vector_type(2))) float v2f;

struct ResTable { float r[L_LV]; };

__global__ __launch_bounds__(256) void hashenc_kernel(
    const float* __restrict__ x,
    const float* __restrict__ ht,
    float* __restrict__ out,
    const int total,            // N * L
    const ResTable res)
{
    const int tid = blockIdx.x * 256 + threadIdx.x;
    if (tid >= total) return;

    const int l = tid & (L_LV - 1);   // level: 16 consecutive lanes share a point
    const int n = tid >> 4;           // point index

    // Point coordinates (16 lanes hit the same 3 dwords -> L1 broadcast).
    const float x0 = x[n * 3 + 0];
    const float x1 = x[n * 3 + 1];
    const float x2 = x[n * 3 + 2];
    const float r  = res.r[l];

    // Scale, floor, fractional weights (matches f32 reference math).
    const float s0 = x0 * r, s1 = x1 * r, s2 = x2 * r;
    const float f0 = floorf(s0), f1 = floorf(s1), f2 = floorf(s2);
    const float w0 = s0 - f0, w1 = s1 - f1, w2 = s2 - f2;
    const unsigned ix = (unsigned)f0;
    const unsigned iy = (unsigned)f1;
    const unsigned iz = (unsigned)f2;

    // Partial hashes: h = (ix+bx)*1 ^ (iy+by)*P1 ^ (iz+bz)*P2, idx = h & (T-1).
    const unsigned hx0 = ix;                const unsigned hx1 = ix + 1u;
    const unsigned hy0 = iy * PRIME1;       const unsigned hy1 = (iy + 1u) * PRIME1;
    const unsigned hz0 = iz * PRIME2;       const unsigned hz1 = (iz + 1u) * PRIME2;

    // Per-level table slice: T float2 entries (4 MB); all 16 slices fit in L2.
    const v2f* __restrict__ tbl = (const v2f*)ht + (size_t)l * T_SZ;

    // Issue all 8 gathers (global_load_b64) before consuming any: max MLP.
    const v2f c000 = tbl[(hx0 ^ hy0 ^ hz0) & T_MASK];
    const v2f c100 = tbl[(hx1 ^ hy0 ^ hz0) & T_MASK];
    const v2f c010 = tbl[(hx0 ^ hy1 ^ hz0) & T_MASK];
    const v2f c110 = tbl[(hx1 ^ hy1 ^ hz0) & T_MASK];
    const v2f c001 = tbl[(hx0 ^ hy0 ^ hz1) & T_MASK];
    const v2f c101 = tbl[(hx1 ^ hy0 ^ hz1) & T_MASK];
    const v2f c011 = tbl[(hx0 ^ hy1 ^ hz1) & T_MASK];
    const v2f c111 = tbl[(hx1 ^ hy1 ^ hz1) & T_MASK];

    // Trilinear weights: corner bit d selects w[d] vs (1 - w[d]).
    const float u0 = 1.0f - w0, u1 = 1.0f - w1, u2 = 1.0f - w2;
    const float wyz00 = u1 * u2;   // (by=0, bz=0)
    const float wyz10 = w1 * u2;   // (by=1, bz=0)
    const float wyz01 = u1 * w2;   // (by=0, bz=1)
    const float wyz11 = w1 * w2;   // (by=1, bz=1)

    v2f acc = c000 * (u0 * wyz00);
    acc += c100 * (w0 * wyz00);
    acc += c010 * (u0 * wyz10);
    acc += c110 * (w0 * wyz10);
    acc += c001 * (u0 * wyz01);
    acc += c101 * (w0 * wyz01);
    acc += c011 * (u0 * wyz11);
    acc += c111 * (w0 * wyz11);

    // Coalesced 8B/lane store; non-temporal so the 64 MB output stream does
    // not evict the L2-resident hash tables (gfx1250 TH=NT cache control).
    __builtin_nontemporal_store(acc, (v2f*)out + tid);
}

extern "C" void kernel_launch(void* const* d_in, const int* in_sizes, int n_in,
                              void* d_out, int out_size, void* d_ws, size_t ws_size,
                              hipStream_t stream) {
    const float* x  = (const float*)d_in[0];   // (N, 3) f32
    const float* ht = (const float*)d_in[1];   // (L, T, F) f32
    float* out      = (float*)d_out;           // (N, L*F) f32

    const int n_pts = in_sizes[0] / 3;
    const int total = n_pts * L_LV;

    // Exactly mirror numpy: b = exp((log(512)-log(16))/15); res_i = floor(16*b^i)
    ResTable res;
    const double b = exp((log(512.0) - log(16.0)) / 15.0);
    for (int i = 0; i < L_LV; ++i)
        res.r[i] = (float)floor(16.0 * pow(b, (double)i));

    const int blocks = (total + 255) / 256;
    hashenc_kernel<<<blocks, 256, 0, stream>>>(x, ht, out, total, res);
}